// MRC_43439299231886
// MI455X (gfx1250) — compile-verified
//
#include <hip/hip_runtime.h>

typedef float v2f __attribute__((ext_vector_type(2)));
typedef float v8f __attribute__((ext_vector_type(8)));
typedef float f4v __attribute__((ext_vector_type(4)));
typedef int   i4v __attribute__((ext_vector_type(4)));

#define B_ 4
#define N_ 1024
#define D_ 768
#define L_ 16
#define EPS_ 1e-7f

// ---- output layout (flat concat, return order) ----
#define SPAN_SZ   (B_*L_*N_*N_)          // 67108864
#define SP_OFF    ((size_t)SPAN_SZ)
#define EP_OFF    (SP_OFF + (size_t)(B_*L_*N_))
#define SCAL_OFF  (EP_OFF + (size_t)(B_*L_*N_))

// ---- workspace layout (bytes) ----
#define ACCF_OFF   0
#define ACCI_OFF   64
#define SEPROB_OFF 256
#define SXT_OFF    (SEPROB_OFF + B_*N_*2*L_*4)   // 256 + 512KB
#define EXT_OFF    (SXT_OFF + B_*L_*N_*4)
#define SPI_OFF    (EXT_OFF + B_*L_*N_*4)
#define EPI_OFF    (SPI_OFF + B_*L_*N_*4)

__device__ inline float sigmoidf(float x) { return 1.0f / (1.0f + __expf(-x)); }

__device__ inline float waveRedF(float v) {
#pragma unroll
  for (int o = 16; o > 0; o >>= 1) v += __shfl_down(v, o, 32);
  return v;
}
__device__ inline int waveRedI(int v) {
#pragma unroll
  for (int o = 16; o > 0; o >>= 1) v += __shfl_down(v, o, 32);
  return v;
}

// ---------------- kernel 0: zero accumulators ----------------
__global__ void k_zero(float* accf, int* acci) {
  int t = threadIdx.x;
  if (t < 8) { accf[t] = 0.0f; acci[t] = 0; }
}

// ---------------- kernel 1: fused WMMA f32 GEMM ----------------
// Y = x @ [w_se | w_sp[:D] | w_sp[D:]]  (4096 x 768 x 64)
// epilogue: cols 0..31  -> sigmoid(+b_se) -> se_prob [B*N, 32]
//           cols 32..47 -> sxT[B,L,N] = val + b_sp[l]
//           cols 48..63 -> exT[B,L,N] = val
__global__ __launch_bounds__(128) void k_gemm(
    const float* __restrict__ x,
    const float* __restrict__ w_se, const float* __restrict__ b_se,
    const float* __restrict__ w_sp, const float* __restrict__ b_sp,
    float* __restrict__ se_prob, float* __restrict__ sxT, float* __restrict__ exT)
{
  const int lane = threadIdx.x & 31;
  const int wave = threadIdx.x >> 5;      // column-tile: 0..3
  const int row0 = blockIdx.x * 16;       // 256 row-tiles over 4096 rows
  const int half = lane >> 4;             // 0 or 1 (K sub-pair select)
  const int nn   = lane & 15;             // A-row (M) and B-col (N) within tile

  const float* wbase; int wstride;
  if (wave == 0)      { wbase = w_se;          wstride = 2 * L_; }
  else if (wave == 1) { wbase = w_se + 16;     wstride = 2 * L_; }
  else if (wave == 2) { wbase = w_sp;          wstride = L_; }
  else                { wbase = w_sp + D_*L_;  wstride = L_; }

  const float* xrow = x + (size_t)(row0 + nn) * D_;

  v8f acc = {};
  for (int k = 0; k < D_; k += 4) {
    const int ka = k + half * 2;
    // A 16x4 f32: lanes 0-15 hold K=k..k+1, lanes 16-31 hold K=k+2..k+3 (M = lane&15)
    v2f a; a.x = xrow[ka]; a.y = xrow[ka + 1];
    // B 4x16 f32: row striped across lanes; same half-split on K
    v2f b; b.x = wbase[ka * wstride + nn];
           b.y = wbase[(ka + 1) * wstride + nn];
    acc = __builtin_amdgcn_wmma_f32_16x16x4_f32(false, a, false, b,
                                                (short)0, acc, false, false);
  }

  const int gcol = wave * 16 + nn;
#pragma unroll
  for (int v = 0; v < 8; ++v) {
    const int m   = v + half * 8;           // C/D layout: VGPR v -> M=v / M=v+8
    const int row = row0 + m;
    const int bb = row >> 10, ii = row & 1023;
    const float val = acc[v];
    if (gcol < 32) {
      se_prob[(size_t)row * 32 + gcol] = sigmoidf(val + b_se[gcol]);
    } else if (gcol < 48) {
      const int l = gcol - 32;
      sxT[((size_t)bb * L_ + l) * N_ + ii] = val + b_sp[l];
    } else {
      const int l = gcol - 48;
      exT[((size_t)bb * L_ + l) * N_ + ii] = val;
    }
  }
}

// ---------------- kernel 2: start/end head ----------------
__global__ __launch_bounds__(256) void k_startend(
    const float* __restrict__ se_prob, const int* __restrict__ start,
    const int* __restrict__ end, const int* __restrict__ seqlen,
    int* __restrict__ spi, int* __restrict__ epi,
    float* __restrict__ out, float* accf)
{
  const int tid = blockIdx.x * 256 + threadIdx.x;   // B*2L*N = 131072
  const int b = tid >> 15;
  const int rem = tid & 32767;
  const int c = rem >> 10;
  const int n = rem & 1023;

  const float p = se_prob[((size_t)b * N_ + n) * 32 + c];
  const int mask = (n < seqlen[b]) ? 1 : 0;
  const int pred = ((p > 0.5f) && mask) ? 1 : 0;
  const int tgt = (c < L_) ? start[((size_t)b * L_ + c) * N_ + n]
                           : end[((size_t)b * L_ + (c - L_)) * N_ + n];
  const float arg  = (tgt == 1) ? (p + EPS_) : (1.0f - p + EPS_);
  const float coef = (tgt == 1) ? (1.0f - p) * (1.0f - p) : p * p;
  const float fl = -0.5f * coef * __logf(arg) * (float)mask;

  if (c < L_) {
    spi[((size_t)b * L_ + c) * N_ + n] = pred;
    out[SP_OFF + ((size_t)b * L_ + c) * N_ + n] = (float)pred;
  } else {
    epi[((size_t)b * L_ + (c - L_)) * N_ + n] = pred;
    out[EP_OFF + ((size_t)b * L_ + (c - L_)) * N_ + n] = (float)pred;
  }

  __shared__ float sred[8];
  const float wv = waveRedF(fl);
  if ((threadIdx.x & 31) == 0) sred[threadIdx.x >> 5] = wv;
  __syncthreads();
  if (threadIdx.x == 0) {
    float s = 0.0f;
#pragma unroll
    for (int i = 0; i < 8; ++i) s += sred[i];
    atomicAdd(accf + 0, s);
  }
}

// ---------------- kernel 3: span head (HBM-bound pass) ----------------
// one block per (b,l,i); threads vectorize over j (4 per thread)
__global__ __launch_bounds__(256) void k_span(
    const float* __restrict__ sxT, const float* __restrict__ exT,
    const int* __restrict__ spi, const int* __restrict__ epi,
    const int* __restrict__ span, const int* __restrict__ val,
    float* __restrict__ out, float* accf, int* acci)
{
  const int bl = blockIdx.x >> 10;                 // b*L + l
  const int i  = blockIdx.x & 1023;
  const size_t rowbase = (size_t)blockIdx.x << 10; // offset of (b,l,i,*) in [B,L,N,N]
  const float slog = sxT[(size_t)bl * N_ + i];     // includes b_sp[l]
  const int sp_i = spi[(size_t)bl * N_ + i];

  const int j0 = threadIdx.x * 4;
  const f4v exv = *(const f4v*)(exT + (size_t)bl * N_ + j0);
  const i4v epv = *(const i4v*)(epi + (size_t)bl * N_ + j0);
  const i4v sv = __builtin_nontemporal_load((const i4v*)(span + rowbase + j0));
  const i4v vv = __builtin_nontemporal_load((const i4v*)(val  + rowbase + j0));

  float fls = 0.0f;
  int vs = 0, as = 0, tps = 0, tns = 0, fps = 0;
  f4v ov;
#pragma unroll
  for (int e = 0; e < 4; ++e) {
    const int j = j0 + e;
    const float logit = slog + exv[e];
    const float p = sigmoidf(logit);
    const int s = sv[e];
    const int v = vv[e];
    const int pred = ((logit > 0.0f) && sp_i && epv[e] && (j >= i)) ? 1 : 0;
    ov[e] = (float)pred;
    const float arg  = (s == 1) ? (p + EPS_) : (1.0f - p + EPS_);
    const float coef = (s == 1) ? (1.0f - p) * (1.0f - p) : p * p;
    fls += -0.5f * coef * __logf(arg) * (float)v;
    vs  += v;
    as  += (pred == s) ? v : 0;
    tps += (s == 1 && pred == 1) ? 1 : 0;
    tns += (s == 1 && pred != 1) ? 1 : 0;
    fps += (s == 0 && pred == 1) ? 1 : 0;
  }
  __builtin_nontemporal_store(ov, (f4v*)(out + rowbase + j0));

  __shared__ float sf[8];
  __shared__ int   si[8][5];
  const float rf = waveRedF(fls);
  const int r0 = waveRedI(vs), r1 = waveRedI(as), r2 = waveRedI(tps),
            r3 = waveRedI(tns), r4 = waveRedI(fps);
  const int w = threadIdx.x >> 5;
  if ((threadIdx.x & 31) == 0) {
    sf[w] = rf; si[w][0] = r0; si[w][1] = r1; si[w][2] = r2; si[w][3] = r3; si[w][4] = r4;
  }
  __syncthreads();
  if (threadIdx.x == 0) {
    float tf = 0.0f; int t0 = 0, t1 = 0, t2 = 0, t3 = 0, t4 = 0;
#pragma unroll
    for (int q = 0; q < 8; ++q) {
      tf += sf[q]; t0 += si[q][0]; t1 += si[q][1]; t2 += si[q][2]; t3 += si[q][3]; t4 += si[q][4];
    }
    atomicAdd(accf + 1, tf);
    atomicAdd(acci + 0, t0); atomicAdd(acci + 1, t1); atomicAdd(acci + 2, t2);
    atomicAdd(acci + 3, t3); atomicAdd(acci + 4, t4);
  }
}

// ---------------- kernel 4: finalize scalars ----------------
__global__ void k_final(const float* accf, const int* acci,
                        const int* __restrict__ seqlen, float* __restrict__ out)
{
  if (threadIdx.x == 0 && blockIdx.x == 0) {
    const float masksum = (float)(2 * L_) *
        (float)(seqlen[0] + seqlen[1] + seqlen[2] + seqlen[3]);
    const float se_loss = accf[0] / masksum;
    const float valsum = (float)acci[0];
    const float span_loss = accf[1] / (valsum + EPS_);
    const float loss = 0.15f * se_loss + 0.7f * span_loss;
    out[SCAL_OFF + 0] = (float)acci[2];  // tp
    out[SCAL_OFF + 1] = (float)acci[3];  // tn (ref's naming; span==1 && pred!=1)
    out[SCAL_OFF + 2] = (float)acci[4];  // fp
    out[SCAL_OFF + 3] = loss;
    out[SCAL_OFF + 4] = (float)acci[1];  // accuracysum
    out[SCAL_OFF + 5] = valsum;
  }
}

extern "C" void kernel_launch(void* const* d_in, const int* in_sizes, int n_in,
                              void* d_out, int out_size, void* d_ws, size_t ws_size,
                              hipStream_t stream)
{
  const float* x      = (const float*)d_in[0];
  const int*   start  = (const int*)d_in[1];
  const int*   end    = (const int*)d_in[2];
  const int*   span   = (const int*)d_in[3];
  const int*   val    = (const int*)d_in[4];
  const int*   seqlen = (const int*)d_in[5];
  const float* w_se   = (const float*)d_in[6];
  const float* b_se   = (const float*)d_in[7];
  const float* w_sp   = (const float*)d_in[8];
  const float* b_sp   = (const float*)d_in[9];

  float* out = (float*)d_out;
  char* ws = (char*)d_ws;
  float* accf    = (float*)(ws + ACCF_OFF);
  int*   acci    = (int*)(ws + ACCI_OFF);
  float* se_prob = (float*)(ws + SEPROB_OFF);
  float* sxT     = (float*)(ws + SXT_OFF);
  float* exT     = (float*)(ws + EXT_OFF);
  int*   spi     = (int*)(ws + SPI_OFF);
  int*   epi     = (int*)(ws + EPI_OFF);

  hipLaunchKernelGGL(k_zero, dim3(1), dim3(64), 0, stream, accf, acci);
  hipLaunchKernelGGL(k_gemm, dim3(256), dim3(128), 0, stream,
                     x, w_se, b_se, w_sp, b_sp, se_prob, sxT, exT);
  hipLaunchKernelGGL(k_startend, dim3(512), dim3(256), 0, stream,
                     se_prob, start, end, seqlen, spi, epi, out, accf);
  hipLaunchKernelGGL(k_span, dim3(B_ * L_ * N_), dim3(256), 0, stream,
                     sxT, exT, spi, epi, span, val, out, accf, acci);
  hipLaunchKernelGGL(k_final, dim3(1), dim3(32), 0, stream, accf, acci, seqlen, out);
}